// GenCTE_22127671509382
// MI455X (gfx1250) — compile-verified
//
#include <hip/hip_runtime.h>
#include <hip/hip_bf16.h>

#define H_DIM   2048
#define U_DIM   1024
#define DU_DIM  256
#define Y_DIM   1024
#define UDU_DIM (U_DIM + DU_DIM)
#define DELTA   0.1f
#define KC      128     // K-chunk staged through LDS (double-buffered)
#define NCHUNK  (H_DIM / KC)
#define ASTRIDE 132     // padded LDS row stride: 132%64=4 -> conflict-free A-frag
                        // reads; 132*4 bytes is 16B-aligned for float4 stores
#define NTERMS  8       // ||M|| ~ 0.08 -> term 8 ~ 1e-13, << fp32 eps

typedef __attribute__((ext_vector_type(2))) float v2f;
typedef __attribute__((ext_vector_type(8))) float v8f;

// ---------------------------------------------------------------------------
// M = 0.5 * DELTA * (W - W^T), tiled transpose through LDS (coalesced R/W).
// ---------------------------------------------------------------------------
__global__ void build_M(const float* __restrict__ W, float* __restrict__ M) {
    __shared__ float Wa[32][33];
    __shared__ float Wb[32][33];
    const int tx = threadIdx.x;           // 0..31
    const int ty = threadIdx.y;           // 0..7
    const int r0 = blockIdx.y * 32;
    const int c0 = blockIdx.x * 32;
    #pragma unroll
    for (int q = 0; q < 4; ++q) {
        const int r = ty + q * 8;
        Wa[r][tx] = W[(size_t)(r0 + r) * H_DIM + c0 + tx];
        Wb[r][tx] = W[(size_t)(c0 + r) * H_DIM + r0 + tx];
    }
    __syncthreads();
    const float s = 0.5f * DELTA;
    #pragma unroll
    for (int q = 0; q < 4; ++q) {
        const int r = ty + q * 8;
        M[(size_t)(r0 + r) * H_DIM + c0 + tx] = s * (Wa[r][tx] - Wb[tx][r]);
    }
}

// ---------------------------------------------------------------------------
// b = B_weight @ [du; u]; init panel P0 = [h, b, 0..0]; hacc = h + DELTA*b.
// One wave32 per output row, float4 dot, shfl reduction.
// ---------------------------------------------------------------------------
__global__ void bvec_init(const float* __restrict__ Bw, const float* __restrict__ du,
                          const float* __restrict__ u,  const float* __restrict__ h,
                          float* __restrict__ P0, float* __restrict__ hacc) {
    __shared__ float udu[UDU_DIM];
    const int tid = threadIdx.x;          // 256 threads = 8 waves
    for (int j = tid; j < UDU_DIM; j += 256)
        udu[j] = (j < DU_DIM) ? du[j] : u[j - DU_DIM];
    __syncthreads();

    const int wave = tid >> 5, lane = tid & 31;
    const int row  = blockIdx.x * 8 + wave;
    const float* Brow = Bw + (size_t)row * UDU_DIM;
    float sum = 0.f;
    #pragma unroll
    for (int it = 0; it < UDU_DIM / 128; ++it) {      // 10 iters
        const int j = (it * 32 + lane) * 4;
        const float4 bv = *(const float4*)(Brow + j);
        sum += bv.x * udu[j] + bv.y * udu[j + 1] + bv.z * udu[j + 2] + bv.w * udu[j + 3];
    }
    #pragma unroll
    for (int off = 16; off; off >>= 1) sum += __shfl_down(sum, off, 32);
    if (lane == 0) {
        const float hv = h[row];
        hacc[row] = hv + DELTA * sum;                 // k = 0 series terms
        float* pr = P0 + row * 16;
        pr[0] = hv;                                   // column 0: h series
        pr[1] = sum;                                  // column 1: b series
        #pragma unroll
        for (int c = 2; c < 16; ++c) pr[c] = 0.f;
    }
}

// ---------------------------------------------------------------------------
// One Taylor step:  Pout = M @ Pin   (2048x2048 * 2048x16, fp32 WMMA),
// then hacc += c0 * Pout[:,0] + c1 * Pout[:,1].
// 128 workgroups x 8 waves. Register-staged, double-buffered LDS pipeline:
// global loads of chunk c+1 are in flight while chunk c computes; dual
// accumulators break the WMMA RAW chain so ds_loads can be hoisted.
// ---------------------------------------------------------------------------
__global__ void gemm_step(const float* __restrict__ M, const float* __restrict__ Pin,
                          float* __restrict__ Pout, float* __restrict__ hacc,
                          float c0, float c1) {
    __shared__ float AsBuf[2][16 * ASTRIDE];   // 16 x KC A-tiles
    __shared__ float PsBuf[2][KC * 16];        // KC x 16 panel tiles
    __shared__ float Red[8 * 8 * 32];          // cross-wave reduction

    const int tid  = threadIdx.x;              // 256 threads = 8 waves
    const int wave = tid >> 5;
    const int lane = tid & 31;
    const int m    = lane & 15;                // A-frag row / D col index
    const int kh   = lane >> 4;                // half-wave select
    const int n    = m;                        // B/D column index
    const int row0 = blockIdx.x * 16;

    // Register staging for one chunk: 4 x float4 per thread.
    const int ra_r0 = tid >> 5,         ra_c0 = (tid & 31) * 4;         // lin4 = tid
    const int ra_r1 = (256 + tid) >> 5, ra_c1 = ((256 + tid) & 31) * 4; // lin4 = 256+tid
    float4 ra0, ra1, rp0, rp1;

    auto load_chunk = [&](int k0) {
        ra0 = *(const float4*)(M + (size_t)(row0 + ra_r0) * H_DIM + k0 + ra_c0);
        ra1 = *(const float4*)(M + (size_t)(row0 + ra_r1) * H_DIM + k0 + ra_c1);
        rp0 = *(const float4*)(Pin + (size_t)k0 * 16 + tid * 4);
        rp1 = *(const float4*)(Pin + (size_t)k0 * 16 + (256 + tid) * 4);
    };
    auto store_chunk = [&](float* As, float* Ps) {
        *(float4*)(As + ra_r0 * ASTRIDE + ra_c0) = ra0;
        *(float4*)(As + ra_r1 * ASTRIDE + ra_c1) = ra1;
        *(float4*)(Ps + tid * 4)         = rp0;
        *(float4*)(Ps + (256 + tid) * 4) = rp1;
    };

    v8f acc0 = {0.f, 0.f, 0.f, 0.f, 0.f, 0.f, 0.f, 0.f};
    v8f acc1 = {0.f, 0.f, 0.f, 0.f, 0.f, 0.f, 0.f, 0.f};

    load_chunk(0);                             // prologue prefetch
    for (int c = 0; c < NCHUNK; ++c) {
        float* As = AsBuf[c & 1];
        float* Ps = PsBuf[c & 1];
        store_chunk(As, Ps);                   // waits on in-flight global loads
        __syncthreads();
        if (c + 1 < NCHUNK) load_chunk((c + 1) * KC);  // overlap with compute

        const int kb = wave * (KC / 8);        // 16 K-values per wave
        #pragma unroll
        for (int kk = 0; kk < KC / 8; kk += 4) {
            const int kc = kb + kk + 2 * kh;
            v2f af, bf;
            af.x = As[m * ASTRIDE + kc];
            af.y = As[m * ASTRIDE + kc + 1];
            bf.x = Ps[(kc)     * 16 + n];
            bf.y = Ps[(kc + 1) * 16 + n];
            if ((kk & 4) == 0)
                acc0 = __builtin_amdgcn_wmma_f32_16x16x4_f32(
                    false, af, false, bf, (short)0, acc0, false, false);
            else
                acc1 = __builtin_amdgcn_wmma_f32_16x16x4_f32(
                    false, af, false, bf, (short)0, acc1, false, false);
        }
        // No trailing barrier needed: next iteration's store targets the other
        // buffer, and every wave passed this iteration's barrier only after
        // finishing the previous chunk's reads of that buffer.
    }

    const v8f acc = acc0 + acc1;

    __syncthreads();
    #pragma unroll
    for (int r = 0; r < 8; ++r) Red[(wave * 8 + r) * 32 + lane] = acc[r];
    __syncthreads();

    if (wave == 0) {
        #pragma unroll
        for (int r = 0; r < 8; ++r) {
            float s = 0.f;
            #pragma unroll
            for (int w = 0; w < 8; ++w) s += Red[(w * 8 + r) * 32 + lane];
            const int mr = row0 + r + 8 * kh;  // D layout: VGPR r -> rows r, r+8
            Pout[mr * 16 + n] = s;
            // col1 value lives one lane over; combine both series contributions.
            const float s1 = __shfl(s, lane + 1, 32);
            if (n == 0) hacc[mr] += c0 * s + c1 * s1;  // unique row per lane
        }
    }
}

// ---------------------------------------------------------------------------
// y = C_weight @ hacc. One wave32 per output row.
// ---------------------------------------------------------------------------
__global__ void out_gemv(const float* __restrict__ Cw, const float* __restrict__ hacc,
                         float* __restrict__ y) {
    __shared__ float hs[H_DIM];
    const int tid = threadIdx.x;          // 256 threads = 8 waves
    for (int j = tid; j < H_DIM; j += 256) hs[j] = hacc[j];
    __syncthreads();

    const int wave = tid >> 5, lane = tid & 31;
    const int row  = blockIdx.x * 8 + wave;
    const float* Crow = Cw + (size_t)row * H_DIM;
    float sum = 0.f;
    #pragma unroll
    for (int it = 0; it < H_DIM / 128; ++it) {        // 16 iters
        const int j = (it * 32 + lane) * 4;
        const float4 cv = *(const float4*)(Crow + j);
        sum += cv.x * hs[j] + cv.y * hs[j + 1] + cv.z * hs[j + 2] + cv.w * hs[j + 3];
    }
    #pragma unroll
    for (int off = 16; off; off >>= 1) sum += __shfl_down(sum, off, 32);
    if (lane == 0) y[row] = sum;
}

// ---------------------------------------------------------------------------
extern "C" void kernel_launch(void* const* d_in, const int* in_sizes, int n_in,
                              void* d_out, int out_size, void* d_ws, size_t ws_size,
                              hipStream_t stream) {
    const float* u  = (const float*)d_in[0];
    const float* du = (const float*)d_in[1];
    const float* W  = (const float*)d_in[2];
    const float* Bw = (const float*)d_in[3];
    const float* Cw = (const float*)d_in[4];
    const float* h0 = (const float*)d_in[5];
    float* y = (float*)d_out;

    float* ws   = (float*)d_ws;
    float* M    = ws;                                 // 2048*2048 fp32 = 16 MB
    float* P0   = M  + (size_t)H_DIM * H_DIM;         // 2048*16
    float* P1   = P0 + (size_t)H_DIM * 16;            // 2048*16
    float* hacc = P1 + (size_t)H_DIM * 16;            // 2048

    build_M<<<dim3(H_DIM / 32, H_DIM / 32), dim3(32, 8), 0, stream>>>(W, M);
    bvec_init<<<H_DIM / 8, 256, 0, stream>>>(Bw, du, u, h0, P0, hacc);

    // hacc += sum_{k=1..NTERMS} [ (-1)^k/k! * M^k h  +  DELTA/(k+1)! * M^k b ]
    double fact = 1.0;
    float* Pa = P0;
    float* Pb = P1;
    for (int k = 1; k <= NTERMS; ++k) {
        fact *= (double)k;
        const float c0 = (float)(((k & 1) ? -1.0 : 1.0) / fact);
        const float c1 = (float)(0.1 / (fact * (double)(k + 1)));
        gemm_step<<<H_DIM / 16, 256, 0, stream>>>(M, Pa, Pb, hacc, c0, c1);
        float* t = Pa; Pa = Pb; Pb = t;
    }

    out_gemv<<<Y_DIM / 8, 256, 0, stream>>>(Cw, hacc, y);
}